// Net_38165079392363
// MI455X (gfx1250) — compile-verified
//
#include <hip/hip_runtime.h>

// ---------------------------------------------------------------------------
// Problem constants (from reference)
// ---------------------------------------------------------------------------
#define T_SEQ   80
#define BATCH   32768
#define IN_F    2
#define HID     10          // hidden size
#define G4H     40          // 4*HID
#define MLP_K   800         // SEQ*HID
#define OUT_F   40          // OUT_LEN*OUT_SIZE

typedef float v2f __attribute__((ext_vector_type(2)));
typedef float v8f __attribute__((ext_vector_type(8)));

// f32 WMMA 16x16x4 : A 16x4 (2 VGPR), B 4x16 (2 VGPR), C/D 16x16 (8 VGPR)
__device__ __forceinline__ v8f wmma4(v2f a, v2f b, v8f c) {
    return __builtin_amdgcn_wmma_f32_16x16x4_f32(
        /*neg_a=*/false, a, /*neg_b=*/false, b,
        /*c_mod=*/(short)0, c, /*reuse_a=*/false, /*reuse_b=*/false);
}

// Branch-free activations using native v_exp_f32 / v_rcp_f32.
__device__ __forceinline__ float sigf(float x) {
    return __builtin_amdgcn_rcpf(1.0f + __expf(-x));
}
__device__ __forceinline__ float tanh_fast(float x) {
    // tanh(x) = sign(x) * (1 - 2/(exp(2|x|)+1)); exp->inf saturates to +/-1.
    float ax = __builtin_fabsf(x);
    float t  = __builtin_fmaf(-2.0f, __builtin_amdgcn_rcpf(1.0f + __expf(2.0f * ax)), 1.0f);
    return __builtin_copysignf(t, x);
}

// ---------------------------------------------------------------------------
// Workspace layout (floats)
// ---------------------------------------------------------------------------
// WB0   : [3][4][2][32]   = 768     layer0 gate weights in WMMA B layout
// WB1   : [5][4][2][32]   = 1280    layer1 gate weights in WMMA B layout
// BIAS0 : [4][16]         = 64      per-gate bias (bih+bhh), lane-broadcast
// BIAS1 : [4][16]         = 64
// WBM   : [200][2][32]    = 12800   MLP W1^T in WMMA B layout
// Y     : [T][B][HID]     = 26214400  tanh(y1) staging
#define WS_WB0   0
#define WS_WB1   768
#define WS_B0    2048
#define WS_B1    2112
#define WS_WBM   2176
#define WS_Y     16384

// ---------------------------------------------------------------------------
// Prep: swizzle weights into WMMA B-matrix layout.
// B layout (32-bit, 4x16): VGPR v, lane l :  K = 4*kstep + v + (l<16 ? 0 : 2),
//                                            N = l & 15
// ---------------------------------------------------------------------------
__global__ void prep_kernel(const float* __restrict__ Wih0, const float* __restrict__ Whh0,
                            const float* __restrict__ bih0, const float* __restrict__ bhh0,
                            const float* __restrict__ Wih1, const float* __restrict__ Whh1,
                            const float* __restrict__ bih1, const float* __restrict__ bhh1,
                            const float* __restrict__ W1,
                            float* __restrict__ ws) {
    float* WB0 = ws + WS_WB0;
    float* WB1 = ws + WS_WB1;
    float* B0  = ws + WS_B0;
    float* B1  = ws + WS_B1;
    float* WBM = ws + WS_WBM;

    const int tid = threadIdx.x + blockIdx.x * blockDim.x;
    const int nt  = blockDim.x * gridDim.x;

    // layer0: concat K = [x(0..1) | h(2..11)], 3 ksteps of 4
    for (int i = tid; i < 3 * 4 * 2 * 32; i += nt) {
        int l = i & 31, v = (i >> 5) & 1, g = (i >> 6) & 3, ks = i >> 8;
        int N = l & 15, K = ks * 4 + v + ((l < 16) ? 0 : 2);
        float val = 0.0f;
        if (N < HID && K < (IN_F + HID)) {
            int n = g * HID + N;
            val = (K < IN_F) ? Wih0[n * IN_F + K] : Whh0[n * HID + (K - IN_F)];
        }
        WB0[i] = val;
    }
    // layer1: concat K = [y0(0..9) | h1(10..19)], 5 ksteps of 4
    for (int i = tid; i < 5 * 4 * 2 * 32; i += nt) {
        int l = i & 31, v = (i >> 5) & 1, g = (i >> 6) & 3, ks = i >> 8;
        int N = l & 15, K = ks * 4 + v + ((l < 16) ? 0 : 2);
        float val = 0.0f;
        if (N < HID && K < 2 * HID) {
            int n = g * HID + N;
            val = (K < HID) ? Wih1[n * HID + K] : Whh1[n * HID + (K - HID)];
        }
        WB1[i] = val;
    }
    // biases per gate-tile, indexed by N (lane), broadcast over M
    for (int i = tid; i < 64; i += nt) {
        int g = i >> 4, N = i & 15;
        B0[i] = (N < HID) ? (bih0[g * HID + N] + bhh0[g * HID + N]) : 0.0f;
        B1[i] = (N < HID) ? (bih1[g * HID + N] + bhh1[g * HID + N]) : 0.0f;
    }
    // MLP: B[k][n] = W1[n][k], 200 ksteps of 4, single N tile (10 of 16 used)
    for (int i = tid; i < 200 * 2 * 32; i += nt) {
        int l = i & 31, v = (i >> 5) & 1, ks = i >> 6;
        int N = l & 15, K = ks * 4 + v + ((l < 16) ? 0 : 2);
        WBM[i] = (N < HID) ? W1[N * MLP_K + K] : 0.0f;
    }
}

// ---------------------------------------------------------------------------
// Fused 2-layer LSTM. One wave per 16-batch tile; h/c live in C-layout
// accumulator registers; gate GEMMs via V_WMMA_F32_16X16X4_F32.
// C/D layout: lane l -> N = l&15 ; VGPR r -> M = r + (l<16 ? 0 : 8).
// A layout:   lane l -> M = l&15 ; VGPR v -> K = 4*ks + v + (l<16 ? 0 : 2).
// ---------------------------------------------------------------------------
__global__ __launch_bounds__(32) void lstm_kernel(const float* __restrict__ x,
                                                  const float* __restrict__ h0_in,
                                                  const float* __restrict__ c0_in,
                                                  const float* __restrict__ ws_w,
                                                  float* __restrict__ Yout) {
    __shared__ float stage[16][24];

    const float* WB0 = ws_w + WS_WB0;
    const float* WB1 = ws_w + WS_WB1;
    const float* B0  = ws_w + WS_B0;
    const float* B1  = ws_w + WS_B1;

    const int lane = threadIdx.x;
    const int N    = lane & 15;
    const int hiK  = (lane < 16) ? 0 : 2;
    const int b0   = blockIdx.x * 16;

    // constant weights into registers (B layout)
    v2f wb0[3][4], wb1[5][4];
    for (int ks = 0; ks < 3; ks++)
        for (int g = 0; g < 4; g++) {
            wb0[ks][g].x = WB0[(((ks * 4 + g) * 2) + 0) * 32 + lane];
            wb0[ks][g].y = WB0[(((ks * 4 + g) * 2) + 1) * 32 + lane];
        }
    for (int ks = 0; ks < 5; ks++)
        for (int g = 0; g < 4; g++) {
            wb1[ks][g].x = WB1[(((ks * 4 + g) * 2) + 0) * 32 + lane];
            wb1[ks][g].y = WB1[(((ks * 4 + g) * 2) + 1) * 32 + lane];
        }
    float bias0[4], bias1[4];
    for (int g = 0; g < 4; g++) { bias0[g] = B0[g * 16 + N]; bias1[g] = B1[g * 16 + N]; }

    // recurrent state in C layout
    v8f h0t, c0t, h1t, c1t;
    for (int r = 0; r < 8; r++) {
        int M = r + ((lane < 16) ? 0 : 8);
        float a = 0.f, b = 0.f, c = 0.f, d = 0.f;
        if (N < HID) {
            size_t i0 = (size_t)(b0 + M) * HID + N;
            a = h0_in[i0];                         // layer 0
            b = c0_in[i0];
            c = h0_in[(size_t)BATCH * HID + i0];   // layer 1
            d = c0_in[(size_t)BATCH * HID + i0];
        }
        h0t[r] = a; c0t[r] = b; h1t[r] = c; c1t[r] = d;
    }

    for (int t = 0; t < T_SEQ; t++) {
        // ---- stage A0 = [ x_t | h0 ]  (16 x 12) ----
        {
            int m = lane >> 1, comp = lane & 1;
            stage[m][comp] = x[((size_t)t * BATCH + b0 + m) * IN_F + comp];
        }
        if (N < HID)
            for (int r = 0; r < 8; r++) {
                int M = r + ((lane < 16) ? 0 : 8);
                stage[M][IN_F + N] = h0t[r];
            }
        __syncthreads();
        v2f a0[3];
        for (int ks = 0; ks < 3; ks++) {
            a0[ks].x = stage[N][ks * 4 + hiK + 0];
            a0[ks].y = stage[N][ks * 4 + hiK + 1];
        }
        __syncthreads();

        // ---- layer0 gates: 3 ksteps x 4 gate tiles ----
        v8f acc[4];
        for (int g = 0; g < 4; g++)
            for (int r = 0; r < 8; r++) acc[g][r] = bias0[g];
        for (int ks = 0; ks < 3; ks++)
            for (int g = 0; g < 4; g++) acc[g] = wmma4(a0[ks], wb0[ks][g], acc[g]);

        for (int r = 0; r < 8; r++) {
            float ig = sigf(acc[0][r]);
            float fg = sigf(acc[1][r]);
            float gg = tanh_fast(acc[2][r]);
            float og = sigf(acc[3][r]);
            float c  = fg * c0t[r] + ig * gg;
            c0t[r] = c;
            h0t[r] = og * tanh_fast(c);
        }

        // ---- stage A1 = [ y0_t | h1 ]  (16 x 20) ----
        if (N < HID)
            for (int r = 0; r < 8; r++) {
                int M = r + ((lane < 16) ? 0 : 8);
                stage[M][N]       = h0t[r];
                stage[M][HID + N] = h1t[r];
            }
        __syncthreads();
        v2f a1[5];
        for (int ks = 0; ks < 5; ks++) {
            a1[ks].x = stage[N][ks * 4 + hiK + 0];
            a1[ks].y = stage[N][ks * 4 + hiK + 1];
        }
        __syncthreads();

        // ---- layer1 gates: 5 ksteps x 4 gate tiles ----
        for (int g = 0; g < 4; g++)
            for (int r = 0; r < 8; r++) acc[g][r] = bias1[g];
        for (int ks = 0; ks < 5; ks++)
            for (int g = 0; g < 4; g++) acc[g] = wmma4(a1[ks], wb1[ks][g], acc[g]);

        for (int r = 0; r < 8; r++) {
            float ig = sigf(acc[0][r]);
            float fg = sigf(acc[1][r]);
            float gg = tanh_fast(acc[2][r]);
            float og = sigf(acc[3][r]);
            float c  = fg * c1t[r] + ig * gg;
            c1t[r] = c;
            h1t[r] = og * tanh_fast(c);
        }

        // ---- emit tanh(y1) for the MLP head ----
        if (N < HID)
            for (int r = 0; r < 8; r++) {
                int M = r + ((lane < 16) ? 0 : 8);
                Yout[((size_t)t * BATCH + b0 + M) * HID + N] = tanh_fast(h1t[r]);
            }
    }
}

// ---------------------------------------------------------------------------
// MLP head: rows are 800 consecutive floats of the flat [T,B,H] tensor
// (faithful to torch .view(batch,-1)). Layer1 (K=800 -> 10) via WMMA,
// layer2 (10 -> 40) scalar via LDS.
// ---------------------------------------------------------------------------
__global__ __launch_bounds__(32) void mlp_kernel(const float* __restrict__ Y,
                                                 const float* __restrict__ ws_w,
                                                 const float* __restrict__ b1,
                                                 const float* __restrict__ W2,
                                                 const float* __restrict__ b2,
                                                 float* __restrict__ out) {
    __shared__ float hv[16][HID];
    const float* WBM = ws_w + WS_WBM;

    const int lane = threadIdx.x;
    const int N    = lane & 15;
    const int hiK  = (lane < 16) ? 0 : 2;
    const int r0   = blockIdx.x * 16;

    float bb = (N < HID) ? b1[N] : 0.0f;
    v8f acc;
    for (int r = 0; r < 8; r++) acc[r] = bb;

    const float* yrow = Y + (size_t)(r0 + N) * MLP_K;  // A: lane = row M
    for (int ks = 0; ks < MLP_K / 4; ks++) {
        v2f a, w;
        a.x = yrow[ks * 4 + hiK + 0];
        a.y = yrow[ks * 4 + hiK + 1];
        w.x = WBM[(ks * 2 + 0) * 32 + lane];
        w.y = WBM[(ks * 2 + 1) * 32 + lane];
        acc = wmma4(a, w, acc);
    }

    if (N < HID)
        for (int r = 0; r < 8; r++) {
            int M = r + ((lane < 16) ? 0 : 8);
            hv[M][N] = sigf(acc[r]);
        }
    __syncthreads();

    // second linear: 16 rows x 40 outputs = 640, strided over 32 lanes
    for (int idx = lane; idx < 16 * OUT_F; idx += 32) {
        int m = idx / OUT_F, o = idx % OUT_F;
        float s = b2[o];
        for (int j = 0; j < HID; j++) s += hv[m][j] * W2[o * HID + j];
        out[(size_t)(r0 + m) * OUT_F + o] = s;
    }
}

// ---------------------------------------------------------------------------
extern "C" void kernel_launch(void* const* d_in, const int* in_sizes, int n_in,
                              void* d_out, int out_size, void* d_ws, size_t ws_size,
                              hipStream_t stream) {
    const float* x    = (const float*)d_in[0];
    const float* h0   = (const float*)d_in[1];
    const float* c0   = (const float*)d_in[2];
    const float* Wih0 = (const float*)d_in[3];
    const float* Whh0 = (const float*)d_in[4];
    const float* bih0 = (const float*)d_in[5];
    const float* bhh0 = (const float*)d_in[6];
    const float* Wih1 = (const float*)d_in[7];
    const float* Whh1 = (const float*)d_in[8];
    const float* bih1 = (const float*)d_in[9];
    const float* bhh1 = (const float*)d_in[10];
    const float* W1   = (const float*)d_in[11];
    const float* b1   = (const float*)d_in[12];
    const float* W2   = (const float*)d_in[13];
    const float* b2   = (const float*)d_in[14];
    float* out = (float*)d_out;
    float* ws  = (float*)d_ws;
    float* Y   = ws + WS_Y;

    prep_kernel<<<8, 256, 0, stream>>>(Wih0, Whh0, bih0, bhh0,
                                       Wih1, Whh1, bih1, bhh1, W1, ws);
    lstm_kernel<<<BATCH / 16, 32, 0, stream>>>(x, h0, c0, ws, Y);
    mlp_kernel<<<BATCH / 16, 32, 0, stream>>>(Y, ws, b1, W2, b2, out);
}